// GlobalEncoder_61864708932179
// MI455X (gfx1250) — compile-verified
//
#include <hip/hip_runtime.h>
#include <hip/hip_bf16.h>
#include <float.h>

#define N_NODES  50000
#define N_EDGES  800000
#define DIM      64
#define N_GRAPHS 64

typedef __attribute__((ext_vector_type(2))) float v2f;
typedef __attribute__((ext_vector_type(8))) float v8f;

// ---------------------------------------------------------------------------
// Kernel 0: initialize workspace accumulators
// ---------------------------------------------------------------------------
__global__ void init_ws_kernel(float* __restrict__ segmax, float* __restrict__ denom,
                               float* __restrict__ sums, float* __restrict__ cnt) {
    int i = blockIdx.x * blockDim.x + threadIdx.x;
    if (i < N_NODES) { segmax[i] = -FLT_MAX; denom[i] = 0.0f; }
    if (i < N_GRAPHS * DIM) sums[i] = 0.0f;
    if (i < N_GRAPHS) cnt[i] = 0.0f;
}

// ---------------------------------------------------------------------------
// One wave computes a 16x16 fp32 output tile of y = x_tile @ W + b via
// 16 chained V_WMMA_F32_16X16X4_F32 (K=64). A-frags come from LDS; B-frags
// come from global (addrspace(1) preserved -> global_load, not flat_load).
//
// A-frag (16x4 f32, ISA 7.12.2): lanes 0-15 hold {K=4k, K=4k+1} for M=lane,
//                                lanes 16-31 hold {K=4k+2, K=4k+3}.
// B-frag (4x16 f32, row-striped): lanes 0-15 hold rows {K=4k, K=4k+1},
//                                 lanes 16-31 hold rows {K=4k+2, K=4k+3}.
// C/D  : VGPR r -> M = r + (lane<16 ? 0 : 8), N = lane&15.
// ---------------------------------------------------------------------------
__device__ __forceinline__ void wmma_tile_16x16(const float* __restrict__ xs,
                                                const float* __restrict__ W,
                                                const float* __restrict__ bia,
                                                float* __restrict__ out,
                                                int m0, int nt, int lane) {
    const int half = lane >> 4;          // 0: K pair {0,1}; 1: K pair {2,3}
    const int l15  = lane & 15;

    v8f c = {};
    #pragma unroll
    for (int kk = 0; kk < 16; ++kk) {
        const int kb = kk * 4 + half * 2;
        v2f a, b;
        a.x = xs[l15 * DIM + kb];
        a.y = xs[l15 * DIM + kb + 1];
        b.x = W[(size_t)kb * DIM + nt + l15];
        b.y = W[(size_t)(kb + 1) * DIM + nt + l15];
        c = __builtin_amdgcn_wmma_f32_16x16x4_f32(
                /*neg_a=*/false, a, /*neg_b=*/false, b,
                /*c_mod=*/(short)0, c, /*reuse_a=*/false, /*reuse_b=*/false);
    }

    const float bval = bia[nt + l15];
    #pragma unroll
    for (int r = 0; r < 8; ++r) {
        const int M = r + half * 8;
        out[(size_t)(m0 + M) * DIM + nt + l15] = c[r] + bval;
    }
}

// ---------------------------------------------------------------------------
// Kernel 1: Q = x@Wq+bq, K = x@Wk+bk, V = x@Wv+bv, agg = x@Ws+bs
// One block = 16 node rows, 512 threads = 16 waves.
// Wave w -> matrix (w/4), 16-column tile (w%4). mat is wave-uniform, so the
// switch keeps EXEC all-ones per wave (WMMA requirement) and keeps each W in
// the global address space.
// ---------------------------------------------------------------------------
__global__ void gemm_qkvs_kernel(const float* __restrict__ x,
                                 const float* __restrict__ Wq, const float* __restrict__ bq,
                                 const float* __restrict__ Wk, const float* __restrict__ bk,
                                 const float* __restrict__ Wv, const float* __restrict__ bv,
                                 const float* __restrict__ Ws, const float* __restrict__ bs,
                                 float* __restrict__ q, float* __restrict__ k,
                                 float* __restrict__ v, float* __restrict__ agg) {
    __shared__ float xs[16 * DIM];                 // 4 KB x-tile shared by 16 waves

    const int m0 = blockIdx.x * 16;

    // prefetch the next block's x tile (global_prefetch_b8)
    if (threadIdx.x < 16 && m0 + 16 < N_NODES)
        __builtin_prefetch(&x[(size_t)(m0 + 16 + threadIdx.x) * DIM], 0, 3);

    // cooperative stage of x[m0:m0+16, 0:64] into LDS
    for (int i = threadIdx.x; i < 16 * DIM; i += blockDim.x) {
        xs[i] = x[(size_t)(m0 + (i >> 6)) * DIM + (i & 63)];
    }
    __syncthreads();

    const int wave = threadIdx.x >> 5;
    const int lane = threadIdx.x & 31;
    const int mat  = wave >> 2;          // 0=Q 1=K 2=V 3=skip
    const int nt   = (wave & 3) * 16;    // output column tile base

    switch (mat) {
        case 0:  wmma_tile_16x16(xs, Wq, bq, q,   m0, nt, lane); break;
        case 1:  wmma_tile_16x16(xs, Wk, bk, k,   m0, nt, lane); break;
        case 2:  wmma_tile_16x16(xs, Wv, bv, v,   m0, nt, lane); break;
        default: wmma_tile_16x16(xs, Ws, bs, agg, m0, nt, lane); break;
    }
}

// ---------------------------------------------------------------------------
// Kernel 2: per-edge attention logits + segment max (wave32 per edge)
// ---------------------------------------------------------------------------
__global__ void edge_logits_kernel(const float* __restrict__ q, const float* __restrict__ k,
                                   const int* __restrict__ ei,
                                   float* __restrict__ segmax, float* __restrict__ logits) {
    const int wid  = (blockIdx.x * blockDim.x + threadIdx.x) >> 5;
    const int lane = threadIdx.x & 31;
    if (wid >= N_EDGES) return;

    const int src = ei[wid];
    const int dst = ei[N_EDGES + wid];

    const float2 qv = ((const float2*)(q + (size_t)dst * DIM))[lane];
    const float2 kv = ((const float2*)(k + (size_t)src * DIM))[lane];
    float s = qv.x * kv.x + qv.y * kv.y;
    #pragma unroll
    for (int off = 16; off > 0; off >>= 1) s += __shfl_xor(s, off, 32);
    s *= 0.125f;                                   // 1/sqrt(64)

    if (lane == 0) {
        logits[wid] = s;
        atomicMax(segmax + dst, s);                // global_atomic_max_num_f32
    }
}

// ---------------------------------------------------------------------------
// Kernel 3: ex = exp(logit - segmax[dst]); denom[dst] += ex  (thread per edge)
// ---------------------------------------------------------------------------
__global__ void edge_exp_kernel(const int* __restrict__ ei, const float* __restrict__ segmax,
                                float* __restrict__ logits, float* __restrict__ denom) {
    const int e = blockIdx.x * blockDim.x + threadIdx.x;
    if (e >= N_EDGES) return;
    const int dst = ei[N_EDGES + e];
    const float ex = __expf(logits[e] - segmax[dst]);
    logits[e] = ex;
    atomicAdd(denom + dst, ex);
}

// ---------------------------------------------------------------------------
// Kernel 4: agg[dst] += (ex/denom[dst]) * v[src]  (wave32 per edge)
// ---------------------------------------------------------------------------
__global__ void edge_scatter_kernel(const float* __restrict__ v, const int* __restrict__ ei,
                                    const float* __restrict__ ex, const float* __restrict__ denom,
                                    float* __restrict__ agg) {
    const int wid  = (blockIdx.x * blockDim.x + threadIdx.x) >> 5;
    const int lane = threadIdx.x & 31;
    if (wid >= N_EDGES) return;

    const int src = ei[wid];
    const int dst = ei[N_EDGES + wid];

    float alpha = 0.0f;
    if (lane == 0) alpha = ex[wid] / denom[dst];
    alpha = __shfl(alpha, 0, 32);

    const float2 vv = ((const float2*)(v + (size_t)src * DIM))[lane];
    float* ap = agg + (size_t)dst * DIM + lane * 2;
    atomicAdd(ap,     alpha * vv.x);               // global_atomic_add_f32
    atomicAdd(ap + 1, alpha * vv.y);
}

// ---------------------------------------------------------------------------
// Kernel 5: out = relu(agg); per-graph sums/counts (wave32 per node)
// ---------------------------------------------------------------------------
__global__ void pool_kernel(const float* __restrict__ agg, const int* __restrict__ batch,
                            float* __restrict__ sums, float* __restrict__ cnt) {
    const int wid  = (blockIdx.x * blockDim.x + threadIdx.x) >> 5;
    const int lane = threadIdx.x & 31;
    if (wid >= N_NODES) return;

    const int g = batch[wid];
    const float2 a = ((const float2*)(agg + (size_t)wid * DIM))[lane];
    float* sp = sums + (size_t)g * DIM + lane * 2;
    atomicAdd(sp,     fmaxf(a.x, 0.0f));
    atomicAdd(sp + 1, fmaxf(a.y, 0.0f));
    if (lane == 0) atomicAdd(cnt + g, 1.0f);
}

// ---------------------------------------------------------------------------
// Kernel 6: out[g,d] = sums[g,d] / max(cnt[g],1)
// ---------------------------------------------------------------------------
__global__ void finalize_kernel(const float* __restrict__ sums, const float* __restrict__ cnt,
                                float* __restrict__ out) {
    const int i = blockIdx.x * blockDim.x + threadIdx.x;
    if (i < N_GRAPHS * DIM) out[i] = sums[i] / fmaxf(cnt[i >> 6], 1.0f);
}

// ---------------------------------------------------------------------------
extern "C" void kernel_launch(void* const* d_in, const int* in_sizes, int n_in,
                              void* d_out, int out_size, void* d_ws, size_t ws_size,
                              hipStream_t stream) {
    (void)in_sizes; (void)n_in; (void)out_size; (void)ws_size;

    const float* x  = (const float*)d_in[0];
    const int*   ei = (const int*)  d_in[1];
    const int*   bt = (const int*)  d_in[2];
    const float* Wq = (const float*)d_in[3];
    const float* bq = (const float*)d_in[4];
    const float* Wk = (const float*)d_in[5];
    const float* bk = (const float*)d_in[6];
    const float* Wv = (const float*)d_in[7];
    const float* bv = (const float*)d_in[8];
    const float* Ws = (const float*)d_in[9];
    const float* bs = (const float*)d_in[10];
    float* out = (float*)d_out;

    // workspace layout (floats)
    float* ws      = (float*)d_ws;
    float* q       = ws;                                   // N*D
    float* k       = q   + (size_t)N_NODES * DIM;          // N*D
    float* v       = k   + (size_t)N_NODES * DIM;          // N*D
    float* agg     = v   + (size_t)N_NODES * DIM;          // N*D (init = x@Ws+bs)
    float* segmax  = agg + (size_t)N_NODES * DIM;          // N
    float* denom   = segmax + N_NODES;                     // N
    float* exbuf   = denom  + N_NODES;                     // E
    float* sums    = exbuf  + N_EDGES;                     // G*D
    float* cnt     = sums   + N_GRAPHS * DIM;              // G

    // 0) init accumulators
    init_ws_kernel<<<(N_NODES + 255) / 256, 256, 0, stream>>>(segmax, denom, sums, cnt);

    // 1) WMMA GEMMs: q/k/v + skip-into-agg   (50000/16 = 3125 tiles)
    gemm_qkvs_kernel<<<N_NODES / 16, 512, 0, stream>>>(
        x, Wq, bq, Wk, bk, Wv, bv, Ws, bs, q, k, v, agg);

    // 2) edge logits + segment max  (wave per edge, 8 waves/block)
    edge_logits_kernel<<<(N_EDGES + 7) / 8, 256, 0, stream>>>(q, k, ei, segmax, exbuf);

    // 3) exp + denominator
    edge_exp_kernel<<<(N_EDGES + 255) / 256, 256, 0, stream>>>(ei, segmax, exbuf, denom);

    // 4) weighted scatter of v into agg
    edge_scatter_kernel<<<(N_EDGES + 7) / 8, 256, 0, stream>>>(v, ei, exbuf, denom, agg);

    // 5) relu + per-graph pooling
    pool_kernel<<<(N_NODES + 7) / 8, 256, 0, stream>>>(agg, bt, sums, cnt);

    // 6) mean
    finalize_kernel<<<(N_GRAPHS * DIM + 255) / 256, 256, 0, stream>>>(sums, cnt, out);
}